// GroupScore_88502096101479
// MI455X (gfx1250) — compile-verified
//
#include <hip/hip_runtime.h>
#include <stdint.h>

// GroupScore for MI455X (gfx1250).
// Memory-bound segmented reduce + elementwise apply. No matmul structure, so
// WMMA is not applicable; the MI455X-relevant optimizations are:
//  - b128 vectorized global loads (2x float4/int4 streams)
//  - LDS-privatized 256-bin histograms via ds atomics (ds_add_f32/ds_min_u32/...)
//  - async global->LDS staging of the parameter table (ASYNCcnt path)
//  - NT-hint output stores so pr+vr (128 MB) stay resident in the 192 MB L2
//    between pass 1 and pass 3 (output is write-once, never re-read).

#define NC      256
#define THREADS 256
#define NBLOCKS 4096

typedef float fvec4 __attribute__((ext_vector_type(4)));

// ---- order-preserving float <-> uint key (for atomic min/max) --------------
__device__ __forceinline__ uint32_t fkey(float x) {
  uint32_t u = __float_as_uint(x);
  return (u & 0x80000000u) ? ~u : (u | 0x80000000u);
}
__device__ __forceinline__ float funkey(uint32_t k) {
  uint32_t u = (k & 0x80000000u) ? (k & 0x7FFFFFFFu) : ~k;
  return __uint_as_float(u);
}

// ---- phase 0: init global accumulators (ws is poisoned, re-init each call) -
__global__ void gs_init(uint32_t* __restrict__ gcnt, float* __restrict__ gsum,
                        uint32_t* __restrict__ gmin, uint32_t* __restrict__ gmax) {
  int t = threadIdx.x;
  gcnt[t] = 0u;
  gsum[t] = 0.0f;
  gmin[t] = 0xFFFFFFFFu;  // identity for unsigned min
  gmax[t] = 0u;           // identity for unsigned max
}

// ---- phase 1: segmented count/sum/min/max via LDS histogram ----------------
__global__ void gs_reduce(const float4* __restrict__ p4, const int4* __restrict__ v4,
                          uint32_t* __restrict__ gcnt, float* __restrict__ gsum,
                          uint32_t* __restrict__ gmin, uint32_t* __restrict__ gmax,
                          int n4) {
  __shared__ uint32_t lcnt[NC];
  __shared__ float    lsum[NC];
  __shared__ uint32_t lmin[NC];
  __shared__ uint32_t lmax[NC];
  const int t = threadIdx.x;
  lcnt[t] = 0u; lsum[t] = 0.0f; lmin[t] = 0xFFFFFFFFu; lmax[t] = 0u;
  __syncthreads();

  const int stride = gridDim.x * blockDim.x;
  for (int i = blockIdx.x * blockDim.x + t; i < n4; i += stride) {
    float4 p = p4[i];            // global_load_b128 (RT: keep resident in L2)
    int4   c = v4[i];            // global_load_b128
    {
      int cc = c.x & (NC - 1); uint32_t k = fkey(p.x);
      atomicAdd(&lcnt[cc], 1u); atomicAdd(&lsum[cc], p.x);   // ds_add_u32 / ds_add_f32
      atomicMin(&lmin[cc], k);  atomicMax(&lmax[cc], k);     // ds_min_u32 / ds_max_u32
    }
    {
      int cc = c.y & (NC - 1); uint32_t k = fkey(p.y);
      atomicAdd(&lcnt[cc], 1u); atomicAdd(&lsum[cc], p.y);
      atomicMin(&lmin[cc], k);  atomicMax(&lmax[cc], k);
    }
    {
      int cc = c.z & (NC - 1); uint32_t k = fkey(p.z);
      atomicAdd(&lcnt[cc], 1u); atomicAdd(&lsum[cc], p.z);
      atomicMin(&lmin[cc], k);  atomicMax(&lmax[cc], k);
    }
    {
      int cc = c.w & (NC - 1); uint32_t k = fkey(p.w);
      atomicAdd(&lcnt[cc], 1u); atomicAdd(&lsum[cc], p.w);
      atomicMin(&lmin[cc], k);  atomicMax(&lmax[cc], k);
    }
  }
  __syncthreads();

  // one global atomic combine per cluster per block
  atomicAdd(&gcnt[t], lcnt[t]);
  atomicAdd(&gsum[t], lsum[t]);
  atomicMin(&gmin[t], lmin[t]);
  atomicMax(&gmax[t], lmax[t]);
}

// ---- phase 2: per-cluster pipeline (single block, 256 threads) -------------
__global__ void gs_params(const uint32_t* __restrict__ gcnt, const float* __restrict__ gsum,
                          const uint32_t* __restrict__ gmin, const uint32_t* __restrict__ gmax,
                          float4* __restrict__ params) {
  __shared__ float svals[NC];
  __shared__ int   sids[NC];
  __shared__ float red[NC];
  const int t = threadIdx.x;

  uint32_t cnt = gcnt[t];
  float mean = gsum[t] / (float)cnt;          // empty cluster: 0/0 = NaN (matches ref)
  bool isn = (mean != mean);
  red[t] = isn ? __builtin_inff() : mean;
  __syncthreads();
  for (int s = NC / 2; s > 0; s >>= 1) {      // min-reduce over valid means
    if (t < s) red[t] = fminf(red[t], red[t + s]);
    __syncthreads();
  }
  float vm = isn ? red[0] * 0.5f : mean;      // NaN means -> valid_min/2

  svals[t] = vm;
  sids[t] = t;
  __syncthreads();

  // bitonic sort ascending, 256 elements, one per thread
  for (int k = 2; k <= NC; k <<= 1) {
    for (int j = k >> 1; j > 0; j >>= 1) {
      int ixj = t ^ j;
      if (ixj > t) {
        float a = svals[t], b = svals[ixj];
        bool up = ((t & k) == 0);
        if ((a > b) == up) {
          svals[t] = b; svals[ixj] = a;
          int s0 = sids[t]; sids[t] = sids[ixj]; sids[ixj] = s0;
        }
      }
      __syncthreads();
    }
  }

  // edge-padded 2-tap smoothing on sorted means (exactly as reference)
  float vmc  = svals[t];
  float prev = (t == 0)      ? vmc        : svals[t - 1];   // left pad: vm_s[0]
  float nxt  = (t == NC - 1) ? vmc * 2.0f : svals[t + 1];   // right pad: 2*vm_s[-1]
  float f0 = (prev + vmc) / 1.99f;
  float f1 = (vmc + nxt) / 2.01f;

  int c = sids[t];  // cluster at sorted position t -> unsort by scatter
  float vmin = funkey(gmin[c]);
  float vmax = funkey(gmax[c]);
  float4 pp;
  if (vmin == vmax) {                         // no_scale substitution
    pp = make_float4(0.0f, 1.0f, 0.0f, 1.0f);
  } else {
    pp = make_float4(vmin, vmax, f0, f1);
  }
  params[c] = pp;
}

// ---- phase 3: elementwise apply --------------------------------------------
__device__ __forceinline__ float apply_one(float p, float4 prm) {
  float tmp = (p - prm.x) / (prm.y - prm.x) * (prm.w - prm.z) + prm.z;
  float sc = p / tmp;
  if (!(__builtin_fabsf(sc) < __builtin_inff())) sc = 0.0f;  // NaN or Inf -> 0
  return p * sc;
}

__global__ void gs_apply(const float4* __restrict__ p4, const int4* __restrict__ v4,
                         const float4* __restrict__ params, float4* __restrict__ o4,
                         int n4) {
  __shared__ float4 sparams[NC];
  const int t = threadIdx.x;

#if defined(__gfx1250__)
  // CDNA5 async global->LDS bulk copy of the 4KB parameter table.
  // One b128 transfer per lane; tracked by ASYNCcnt (CDNA5 ISA).
  {
    uint32_t lds_off = (uint32_t)(uintptr_t)(__attribute__((address_space(3))) float4*)&sparams[t];
    uint64_t gaddr = (uint64_t)(uintptr_t)(params + t);
    asm volatile("global_load_async_to_lds_b128 %0, %1, off"
                 :: "v"(lds_off), "v"(gaddr) : "memory");
    asm volatile("s_wait_asynccnt 0x0" ::: "memory");
  }
#else
  sparams[t] = params[t];
#endif
  __syncthreads();

  const int stride = gridDim.x * blockDim.x;
  for (int i = blockIdx.x * blockDim.x + t; i < n4; i += stride) {
    float4 p = p4[i];   // expected L2 hit: pr stayed resident since pass 1
    int4   c = v4[i];   // expected L2 hit
    float4 o;
    o.x = apply_one(p.x, sparams[c.x & (NC - 1)]);
    o.y = apply_one(p.y, sparams[c.y & (NC - 1)]);
    o.z = apply_one(p.z, sparams[c.z & (NC - 1)]);
    o.w = apply_one(p.w, sparams[c.w & (NC - 1)]);
    // NT-hint b128 store: output is write-once, never re-read; keep it from
    // evicting pr/vr out of the 192 MB L2 (CDNA5 ISA store TH=NT).
    fvec4 ov = {o.x, o.y, o.z, o.w};
    __builtin_nontemporal_store(ov, (fvec4*)&o4[i]);
  }
}

// ---- launcher ---------------------------------------------------------------
extern "C" void kernel_launch(void* const* d_in, const int* in_sizes, int n_in,
                              void* d_out, int out_size, void* d_ws, size_t ws_size,
                              hipStream_t stream) {
  const float* pr = (const float*)d_in[0];
  const int*   vr = (const int*)d_in[1];
  float* out = (float*)d_out;
  const int N  = in_sizes[0];   // 1024 * 16384 = 16,777,216 (divisible by 4)
  const int n4 = N / 4;

  uint8_t* ws = (uint8_t*)d_ws;
  uint32_t* gcnt   = (uint32_t*)(ws);
  float*    gsum   = (float*)   (ws + NC * 4);
  uint32_t* gmin   = (uint32_t*)(ws + NC * 8);
  uint32_t* gmax   = (uint32_t*)(ws + NC * 12);
  float4*   params = (float4*)  (ws + NC * 16);   // 8 KB total workspace use

  gs_init  <<<1, NC, 0, stream>>>(gcnt, gsum, gmin, gmax);
  gs_reduce<<<NBLOCKS, THREADS, 0, stream>>>((const float4*)pr, (const int4*)vr,
                                             gcnt, gsum, gmin, gmax, n4);
  gs_params<<<1, NC, 0, stream>>>(gcnt, gsum, gmin, gmax, params);
  gs_apply <<<NBLOCKS, THREADS, 0, stream>>>((const float4*)pr, (const int4*)vr,
                                             params, (float4*)out, n4);
}